// Net_2723009265794
// MI455X (gfx1250) — compile-verified
//
#include <hip/hip_runtime.h>
#include <math.h>

#define BN_EPS   1e-5f
#define NEG_INF  (-3.402823466e+38f)
#define LDA      132   // LDS stride (floats) for 16x128 A strip: (m*4+k)%64 banks conflict-free

typedef __attribute__((ext_vector_type(2))) float v2f;
typedef __attribute__((ext_vector_type(8))) float v8f;

// ---------------------------------------------------------------- utilities
static __device__ __forceinline__ float warp_sum(float v) {
#pragma unroll
  for (int o = 16; o > 0; o >>= 1) v += __shfl_xor(v, o, 32);
  return v;
}
static __device__ __forceinline__ float warp_max(float v) {
#pragma unroll
  for (int o = 16; o > 0; o >>= 1) v = fmaxf(v, __shfl_xor(v, o, 32));
  return v;
}
static __device__ __forceinline__ void atomic_max_float(float* addr, float value) {
  // order-preserving int trick; init value must be -FLT_MAX
  if (value >= 0.f) atomicMax((int*)addr, __float_as_int(value));
  else              atomicMin((unsigned int*)addr, __float_as_uint(value));
}

__global__ void k_fill(float* __restrict__ p, float v, long long n) {
  long long i = (long long)blockIdx.x * blockDim.x + threadIdx.x;
  if (i < n) p[i] = v;
}

// ---------------------------------------------------------------- weight prep
// WP[K x NoutPad] = s * W[K x Nout], zero-padded columns; s = cw[cwidx]/sum(cw) or 1
__global__ void k_prepw(const float* __restrict__ W, float* __restrict__ WP,
                        int Nout, int NoutPad, int total, const float* __restrict__ cw,
                        int cwidx) {
  int idx = blockIdx.x * blockDim.x + threadIdx.x;
  if (idx >= total) return;
  int col = idx % NoutPad;
  int row = idx / NoutPad;
  float s = 1.f;
  if (cw) s = cw[cwidx] / (cw[0] + cw[1] + cw[2]);
  WP[idx] = (col < Nout) ? s * W[row * Nout + col] : 0.f;
}

// ---------------------------------------------------------------- degrees / norms
__global__ void k_deg(const int* __restrict__ src, const int* __restrict__ dst,
                      float* __restrict__ indeg, float* __restrict__ outdeg, int E) {
  int e = blockIdx.x * blockDim.x + threadIdx.x;
  if (e >= E) return;
  atomicAdd(&indeg[dst[e]], 1.f);
  atomicAdd(&outdeg[src[e]], 1.f);
}

__global__ void k_norms(const float* __restrict__ indeg, const float* __restrict__ outdeg,
                        float* __restrict__ ns, float* __restrict__ nd, int N) {
  int i = blockIdx.x * blockDim.x + threadIdx.x;
  if (i >= N) return;
  ns[i] = rsqrtf(fmaxf(outdeg[i], 1.f));
  nd[i] = rsqrtf(fmaxf(indeg[i], 1.f));
}

// -------------------------------------------------- fused SAGE-sum + GCN-scaled-sum gather
// one wave per edge, lane = float4 chunk of the 128-float row
__global__ __launch_bounds__(256)
void k_gather128(const float* __restrict__ X, const int* __restrict__ src,
                 const int* __restrict__ dst, const float* __restrict__ ns,
                 float* __restrict__ neis, float* __restrict__ gcns, int E) {
  long long gid = (long long)blockIdx.x * blockDim.x + threadIdx.x;
  int e = (int)(gid >> 5);
  if (e >= E) return;
  int lane = (int)(gid & 31);
  int s = src[e], d = dst[e];
  float sc = ns[s];
  const float4 v = *(const float4*)(X + s * 128 + lane * 4);
  float* np = neis + d * 128 + lane * 4;
  float* gp = gcns + d * 128 + lane * 4;
  atomicAdd(np + 0, v.x); atomicAdd(np + 1, v.y);
  atomicAdd(np + 2, v.z); atomicAdd(np + 3, v.w);
  atomicAdd(gp + 0, sc * v.x); atomicAdd(gp + 1, sc * v.y);
  atomicAdd(gp + 2, sc * v.z); atomicAdd(gp + 3, sc * v.w);
}

// nei /= max(indeg,1);  gcn *= nd
__global__ void k_postagg128(float* __restrict__ neis, float* __restrict__ gcns,
                             const float* __restrict__ indeg, const float* __restrict__ nd,
                             int total) {
  int idx = blockIdx.x * blockDim.x + threadIdx.x;
  if (idx >= total) return;
  int i = idx >> 7;
  neis[idx] *= 1.f / fmaxf(indeg[i], 1.f);
  gcns[idx] *= nd[i];
}

// ---------------------------------------------------------------- WMMA fp32 GEMM core
static __device__ __forceinline__ void stage_strip(const float* __restrict__ A, int m0,
                                                   float* __restrict__ As) {
  // 128 threads stage 16 rows x 128 cols into LDS (stride LDA)
  int t = threadIdx.x;
#pragma unroll
  for (int j = 0; j < 4; ++j) {
    int q = t + j * 128;          // float4 index 0..511
    int row = q >> 5;
    int c4 = (q & 31) << 2;
    *(float4*)(&As[row * LDA + c4]) = *(const float4*)(A + (m0 + row) * 128 + c4);
  }
}

// one 16x16 tile, K=128 via 32x v_wmma_f32_16x16x4_f32; W is pre-padded to NPAD cols
template <int NPAD>
static __device__ __forceinline__ v8f tile_acc(const float* __restrict__ As,
                                               const float* __restrict__ W, int n0,
                                               int lane, v8f acc) {
  const int row   = lane & 15;
  const int khalf = (lane >> 4) << 1;                 // 0 or 2
  const float* wp = W + khalf * NPAD + n0 + (lane & 15);
  const float* ap = &As[row * LDA + khalf];
#pragma unroll
  for (int kb = 0; kb < 32; ++kb) {
    v2f a = *(const v2f*)(ap + kb * 4);               // A(m=row, k..k+1), ds imm offset
    v2f b;
    b.x = wp[kb * 4 * NPAD];                          // B(k,   n), global imm offset
    b.y = wp[kb * 4 * NPAD + NPAD];                   // B(k+1, n)
    acc = __builtin_amdgcn_wmma_f32_16x16x4_f32(false, a, false, b, (short)0, acc,
                                                false, false);
  }
  return acc;
}

// out = A @ W   (K=128, W padded to NPAD cols), used for GAT feature projection
template <int NPAD>
__global__ __launch_bounds__(128)
void k_gemm1(const float* __restrict__ A, const float* __restrict__ W,
             float* __restrict__ out, int ldout, int Nout) {
  constexpr int NTILES = NPAD / 16;
  constexpr int TPW    = (NTILES + 3) / 4;            // tiles per wave (max)
  __shared__ float As[16 * LDA];
  const int m0 = blockIdx.x * 16;
  const int lane = threadIdx.x & 31;
  const int wave = threadIdx.x >> 5;
  stage_strip(A, m0, As);
  __syncthreads();
  v8f tot[TPW];
#pragma unroll
  for (int ti = 0; ti < TPW; ++ti) tot[ti] = {};
#pragma unroll
  for (int ti = 0; ti < TPW; ++ti) {
    int nt = wave + ti * 4;
    if (nt < NTILES) tot[ti] = tile_acc<NPAD>(As, W, nt * 16, lane, tot[ti]);
  }
#pragma unroll
  for (int ti = 0; ti < TPW; ++ti) {
    int nt = wave + ti * 4;
    int n = nt * 16 + (lane & 15);
    if (nt < NTILES && n < Nout) {
#pragma unroll
      for (int j = 0; j < 8; ++j) {
        int m = m0 + j + ((lane >> 4) << 3);
        out[m * ldout + n] = tot[ti][j];
      }
    }
  }
}

// out = (A0@W0p) + (A1@W1p) + (A2@W2p) + w2*x3 + (w0*b0 + w1*b1 + w2*b2)
// W*p are pre-scaled (w0,w0,w1) and pre-padded to NPAD cols -> single acc chain.
template <int NPAD>
__global__ __launch_bounds__(128)
void k_gemm3(const float* __restrict__ A0, const float* __restrict__ A1,
             const float* __restrict__ A2, const float* __restrict__ W0p,
             const float* __restrict__ W1p, const float* __restrict__ W2p,
             const float* __restrict__ b0, const float* __restrict__ b1,
             const float* __restrict__ b2, const float* __restrict__ x3, int ldx3,
             const float* __restrict__ cw, float* __restrict__ out, int ldout, int Nout) {
  constexpr int NTILES = NPAD / 16;
  constexpr int TPW    = (NTILES + 3) / 4;
  __shared__ float As[16 * LDA];
  const int m0 = blockIdx.x * 16;
  const int lane = threadIdx.x & 31;
  const int wave = threadIdx.x >> 5;

  const float csum = cw[0] + cw[1] + cw[2];
  const float w0 = cw[0] / csum, w1 = cw[1] / csum, w2 = cw[2] / csum;
  const float* Am[3] = {A0, A1, A2};
  const float* Wm[3] = {W0p, W1p, W2p};

  v8f tot[TPW];
#pragma unroll
  for (int ti = 0; ti < TPW; ++ti) tot[ti] = {};

#pragma unroll
  for (int term = 0; term < 3; ++term) {
    __syncthreads();
    stage_strip(Am[term], m0, As);
    __syncthreads();
#pragma unroll
    for (int ti = 0; ti < TPW; ++ti) {
      int nt = wave + ti * 4;
      if (nt < NTILES) tot[ti] = tile_acc<NPAD>(As, Wm[term], nt * 16, lane, tot[ti]);
    }
  }

#pragma unroll
  for (int ti = 0; ti < TPW; ++ti) {
    int nt = wave + ti * 4;
    int n = nt * 16 + (lane & 15);
    if (nt < NTILES && n < Nout) {
      float bias = w0 * b0[n] + w1 * b1[n] + w2 * b2[n];
#pragma unroll
      for (int j = 0; j < 8; ++j) {
        int m = m0 + j + ((lane >> 4) << 3);
        out[m * ldout + n] = tot[ti][j] + w2 * x3[m * ldx3 + n] + bias;
      }
    }
  }
}

// ---------------------------------------------------------------- GAT pieces
// el[i] = f[i,:]·al , er[i] = f[i,:]·ar     (128-wide, one wave per node)
__global__ __launch_bounds__(256)
void k_dots128(const float* __restrict__ F, const float* __restrict__ al,
               const float* __restrict__ ar, float* __restrict__ el,
               float* __restrict__ er, int N) {
  long long gid = (long long)blockIdx.x * blockDim.x + threadIdx.x;
  int i = (int)(gid >> 5);
  if (i >= N) return;
  int lane = (int)(gid & 31);
  float4 f = *(const float4*)(F + i * 128 + lane * 4);
  float4 a = *(const float4*)(al + lane * 4);
  float4 b = *(const float4*)(ar + lane * 4);
  float pl = f.x * a.x + f.y * a.y + f.z * a.z + f.w * a.w;
  float pr = f.x * b.x + f.y * b.y + f.z * b.z + f.w * b.w;
  pl = warp_sum(pl);
  pr = warp_sum(pr);
  if (lane == 0) { el[i] = pl; er[i] = pr; }
}

// 40-wide variant (F stride ld, 20 lanes x float2)
__global__ __launch_bounds__(256)
void k_dots40(const float* __restrict__ F, const float* __restrict__ al,
              const float* __restrict__ ar, float* __restrict__ el,
              float* __restrict__ er, int N, int ld) {
  long long gid = (long long)blockIdx.x * blockDim.x + threadIdx.x;
  int i = (int)(gid >> 5);
  if (i >= N) return;
  int lane = (int)(gid & 31);
  float pl = 0.f, pr = 0.f;
  if (lane < 20) {
    float2 f = *(const float2*)(F + i * ld + lane * 2);
    float2 a = *(const float2*)(al + lane * 2);
    float2 b = *(const float2*)(ar + lane * 2);
    pl = f.x * a.x + f.y * a.y;
    pr = f.x * b.x + f.y * b.y;
  }
  pl = warp_sum(pl);
  pr = warp_sum(pr);
  if (lane == 0) { el[i] = pl; er[i] = pr; }
}

__global__ void k_edge_max(const float* __restrict__ el, const float* __restrict__ er,
                           const int* __restrict__ src, const int* __restrict__ dst,
                           float* __restrict__ earr, float* __restrict__ m, int E) {
  int e = blockIdx.x * blockDim.x + threadIdx.x;
  if (e >= E) return;
  float v = el[src[e]] + er[dst[e]];
  v = v > 0.f ? v : 0.2f * v;                 // leaky_relu(0.2)
  earr[e] = v;
  atomic_max_float(&m[dst[e]], v);
}

__global__ void k_edge_exp(float* __restrict__ earr, const float* __restrict__ m,
                           const int* __restrict__ dst, float* __restrict__ s, int E) {
  int e = blockIdx.x * blockDim.x + threadIdx.x;
  if (e >= E) return;
  float a = expf(earr[e] - m[dst[e]]);
  earr[e] = a;
  atomicAdd(&s[dst[e]], a);
}

// x3[dst] += alpha * f[src]   (128-wide rows)
__global__ __launch_bounds__(256)
void k_gatherF128(const float* __restrict__ F, const float* __restrict__ earr,
                  const float* __restrict__ ssum, const int* __restrict__ src,
                  const int* __restrict__ dst, float* __restrict__ x3, int E) {
  long long gid = (long long)blockIdx.x * blockDim.x + threadIdx.x;
  int e = (int)(gid >> 5);
  if (e >= E) return;
  int lane = (int)(gid & 31);
  int s = src[e], d = dst[e];
  float alpha = earr[e] / fmaxf(ssum[d], 1e-9f);
  const float4 v = *(const float4*)(F + s * 128 + lane * 4);
  float* xp = x3 + d * 128 + lane * 4;
  atomicAdd(xp + 0, alpha * v.x); atomicAdd(xp + 1, alpha * v.y);
  atomicAdd(xp + 2, alpha * v.z); atomicAdd(xp + 3, alpha * v.w);
}

// 40-wide variant, rows stride ld
__global__ __launch_bounds__(256)
void k_gatherF40(const float* __restrict__ F, const float* __restrict__ earr,
                 const float* __restrict__ ssum, const int* __restrict__ src,
                 const int* __restrict__ dst, float* __restrict__ x3, int E, int ld) {
  long long gid = (long long)blockIdx.x * blockDim.x + threadIdx.x;
  int e = (int)(gid >> 5);
  if (e >= E) return;
  int lane = (int)(gid & 31);
  if (lane >= 20) return;
  int s = src[e], d = dst[e];
  float alpha = earr[e] / fmaxf(ssum[d], 1e-9f);
  float2 v = *(const float2*)(F + s * ld + lane * 2);
  float* xp = x3 + d * ld + lane * 2;
  atomicAdd(xp + 0, alpha * v.x);
  atomicAdd(xp + 1, alpha * v.y);
}

// ---------------------------------------------------------------- BatchNorm
__global__ __launch_bounds__(128)
void k_bnstat(const float* __restrict__ h, float* __restrict__ colsum,
              float* __restrict__ colsq, int N) {
  int c = threadIdx.x;                  // 0..127
  int r0 = blockIdx.x * 256;
  int r1 = min(r0 + 256, N);
  float s = 0.f, q = 0.f;
  for (int r = r0; r < r1; ++r) {
    float v = h[r * 128 + c];
    s += v; q += v * v;
  }
  atomicAdd(&colsum[c], s);
  atomicAdd(&colsq[c], q);
}

__global__ void k_bnapply(float* __restrict__ h, const float* __restrict__ colsum,
                          const float* __restrict__ colsq, const float* __restrict__ g,
                          const float* __restrict__ be, float invN, int total) {
  int idx = blockIdx.x * blockDim.x + threadIdx.x;
  if (idx >= total) return;
  int c = idx & 127;
  float mu = colsum[c] * invN;
  float var = colsq[c] * invN - mu * mu;
  float v = (h[idx] - mu) * rsqrtf(var + BN_EPS) * g[c] + be[c];
  h[idx] = fmaxf(v, 0.f);
}

// ---------------------------------------------------------------- log_softmax (40 classes)
__global__ __launch_bounds__(256)
void k_logsoftmax(const float* __restrict__ h2, float* __restrict__ out, int N, int ld) {
  long long gid = (long long)blockIdx.x * blockDim.x + threadIdx.x;
  int i = (int)(gid >> 5);
  if (i >= N) return;
  int lane = (int)(gid & 31);
  float2 v = make_float2(NEG_INF, NEG_INF);
  if (lane < 20) v = *(const float2*)(h2 + i * ld + lane * 2);
  float mx = warp_max(fmaxf(v.x, v.y));
  float es = (lane < 20) ? (expf(v.x - mx) + expf(v.y - mx)) : 0.f;
  es = warp_sum(es);
  float lse = mx + logf(es);
  if (lane < 20) {
    out[i * 40 + lane * 2 + 0] = v.x - lse;
    out[i * 40 + lane * 2 + 1] = v.y - lse;
  }
}

// ---------------------------------------------------------------- host launcher
static inline unsigned cdivu(long long a, long long b) { return (unsigned)((a + b - 1) / b); }

extern "C" void kernel_launch(void* const* d_in, const int* in_sizes, int n_in,
                              void* d_out, int out_size, void* d_ws, size_t ws_size,
                              hipStream_t stream) {
  const float* x    = (const float*)d_in[0];
  const int*   src  = (const int*)d_in[1];
  const int*   dst  = (const int*)d_in[2];
  const float* w11s = (const float*)d_in[3];
  const float* w11n = (const float*)d_in[4];
  const float* b11  = (const float*)d_in[5];
  const float* w12  = (const float*)d_in[6];
  const float* b12  = (const float*)d_in[7];
  const float* w13  = (const float*)d_in[8];
  const float* a13l = (const float*)d_in[9];
  const float* a13r = (const float*)d_in[10];
  const float* b13  = (const float*)d_in[11];
  const float* cw1  = (const float*)d_in[12];
  const float* g    = (const float*)d_in[13];
  const float* be   = (const float*)d_in[14];
  const float* w21s = (const float*)d_in[15];
  const float* w21n = (const float*)d_in[16];
  const float* b21  = (const float*)d_in[17];
  const float* w22  = (const float*)d_in[18];
  const float* b22  = (const float*)d_in[19];
  const float* w23  = (const float*)d_in[20];
  const float* a23l = (const float*)d_in[21];
  const float* a23r = (const float*)d_in[22];
  const float* b23  = (const float*)d_in[23];
  const float* cw2  = (const float*)d_in[24];

  const int N  = in_sizes[0] / 128;
  const int E  = in_sizes[1];
  const int DC = in_sizes[17];   // 40
  const int LD2 = 48;            // padded stride for 40-wide temps
  float* out = (float*)d_out;

  // ---- workspace carve (all region sizes multiples of 4 floats -> 16B aligned)
  float* w = (float*)d_ws;
  size_t off = 0;
  auto carve = [&](size_t n) { float* p = w + off; off += n; return p; };
  float* indeg  = carve(N);
  float* outdeg = carve(N);
  float* ns     = carve(N);
  float* nd     = carve(N);
  float* mbuf   = carve(N);
  float* sbuf   = carve(N);
  float* el     = carve(N);
  float* er     = carve(N);
  float* colsum = carve(128);
  float* colsq  = carve(128);
  float* wp11s  = carve(128 * 128);   // pre-scaled (w0)
  float* wp11n  = carve(128 * 128);   // pre-scaled (w0)
  float* wp12   = carve(128 * 128);   // pre-scaled (w1)
  float* wp21s  = carve(128 * 48);    // pre-scaled (w0), padded
  float* wp21n  = carve(128 * 48);    // pre-scaled (w0), padded
  float* wp22   = carve(128 * 48);    // pre-scaled (w1), padded
  float* wp23   = carve(128 * 48);    // unscaled, padded
  float* nei    = carve((size_t)N * 128);
  float* gcn    = carve((size_t)N * 128);
  float* fbuf   = carve((size_t)N * 128);   // reused as 48-stride f2 in layer 2
  float* x3     = carve((size_t)N * 128);   // reused as 48-stride x3 in layer 2
  float* hbuf   = carve((size_t)N * 128);   // h_pre -> h (in-place BN+ReLU)
  float* h2     = carve((size_t)N * LD2);
  float* earr   = carve(E);

  const long long NF = (long long)N * 128;
  const unsigned TB = 256;
  const unsigned gN    = cdivu(N, TB);
  const unsigned gE    = cdivu(E, TB);
  const unsigned gEw   = cdivu((long long)E * 32, TB);   // wave-per-edge
  const unsigned gNw   = cdivu((long long)N * 32, TB);   // wave-per-node
  const unsigned gNF   = cdivu(NF, TB);
  const unsigned gGemm = (unsigned)(N / 16);             // N divisible by 16

  // ---- weight prep: scale by normalized cw, pad to NPAD columns
  k_prepw<<<64, TB, 0, stream>>>(w11s, wp11s, 128, 128, 128 * 128, cw1, 0);
  k_prepw<<<64, TB, 0, stream>>>(w11n, wp11n, 128, 128, 128 * 128, cw1, 0);
  k_prepw<<<64, TB, 0, stream>>>(w12,  wp12,  128, 128, 128 * 128, cw1, 1);
  k_prepw<<<24, TB, 0, stream>>>(w21s, wp21s, DC, 48, 128 * 48, cw2, 0);
  k_prepw<<<24, TB, 0, stream>>>(w21n, wp21n, DC, 48, 128 * 48, cw2, 0);
  k_prepw<<<24, TB, 0, stream>>>(w22,  wp22,  DC, 48, 128 * 48, cw2, 1);
  k_prepw<<<24, TB, 0, stream>>>(w23,  wp23,  DC, 48, 128 * 48, nullptr, 0);

  // =========================== Layer 1 ===========================
  k_fill<<<cdivu(2LL * N, TB), TB, 0, stream>>>(indeg, 0.f, 2LL * N);
  k_deg<<<gE, TB, 0, stream>>>(src, dst, indeg, outdeg, E);
  k_norms<<<gN, TB, 0, stream>>>(indeg, outdeg, ns, nd, N);

  k_fill<<<cdivu(2 * NF, TB), TB, 0, stream>>>(nei, 0.f, 2 * NF);
  k_gather128<<<gEw, TB, 0, stream>>>(x, src, dst, ns, nei, gcn, E);
  k_postagg128<<<gNF, TB, 0, stream>>>(nei, gcn, indeg, nd, (int)NF);

  // GAT: f = x @ w13, attention logits, edge softmax, aggregate
  k_gemm1<128><<<gGemm, 128, 0, stream>>>(x, w13, fbuf, 128, 128);
  k_dots128<<<gNw, TB, 0, stream>>>(fbuf, a13l, a13r, el, er, N);
  k_fill<<<gN, TB, 0, stream>>>(mbuf, NEG_INF, N);
  k_fill<<<gN, TB, 0, stream>>>(sbuf, 0.f, N);
  k_edge_max<<<gE, TB, 0, stream>>>(el, er, src, dst, earr, mbuf, E);
  k_edge_exp<<<gE, TB, 0, stream>>>(earr, mbuf, dst, sbuf, E);
  k_fill<<<gNF, TB, 0, stream>>>(x3, 0.f, NF);
  k_gatherF128<<<gEw, TB, 0, stream>>>(fbuf, earr, sbuf, src, dst, x3, E);

  // fused combine: h_pre = x@W0p + nei@W1p + gcn@W2p + w2*x3 + bias
  k_gemm3<128><<<gGemm, 128, 0, stream>>>(x, nei, gcn, wp11s, wp11n, wp12,
                                          b11, b12, b13, x3, 128, cw1, hbuf, 128, 128);

  // BatchNorm (batch stats) + ReLU, in place
  k_fill<<<1, TB, 0, stream>>>(colsum, 0.f, 256);
  k_bnstat<<<cdivu(N, 256), 128, 0, stream>>>(hbuf, colsum, colsq, N);
  k_bnapply<<<gNF, TB, 0, stream>>>(hbuf, colsum, colsq, g, be, 1.f / (float)N, (int)NF);

  // =========================== Layer 2 ===========================
  k_fill<<<cdivu(2 * NF, TB), TB, 0, stream>>>(nei, 0.f, 2 * NF);
  k_gather128<<<gEw, TB, 0, stream>>>(hbuf, src, dst, ns, nei, gcn, E);
  k_postagg128<<<gNF, TB, 0, stream>>>(nei, gcn, indeg, nd, (int)NF);

  k_gemm1<48><<<gGemm, 128, 0, stream>>>(hbuf, wp23, fbuf, LD2, DC);
  k_dots40<<<gNw, TB, 0, stream>>>(fbuf, a23l, a23r, el, er, N, LD2);
  k_fill<<<gN, TB, 0, stream>>>(mbuf, NEG_INF, N);
  k_fill<<<gN, TB, 0, stream>>>(sbuf, 0.f, N);
  k_edge_max<<<gE, TB, 0, stream>>>(el, er, src, dst, earr, mbuf, E);
  k_edge_exp<<<gE, TB, 0, stream>>>(earr, mbuf, dst, sbuf, E);
  k_fill<<<cdivu((long long)N * LD2, TB), TB, 0, stream>>>(x3, 0.f, (long long)N * LD2);
  k_gatherF40<<<gEw, TB, 0, stream>>>(fbuf, earr, sbuf, src, dst, x3, E, LD2);

  k_gemm3<48><<<gGemm, 128, 0, stream>>>(hbuf, nei, gcn, wp21s, wp21n, wp22,
                                         b21, b22, b23, x3, LD2, cw2, h2, LD2, DC);

  k_logsoftmax<<<gNw, TB, 0, stream>>>(h2, out, N, LD2);
}